// MPNN_4269197492601
// MI455X (gfx1250) — compile-verified
//
#include <hip/hip_runtime.h>

typedef __attribute__((ext_vector_type(2))) float v2f;
typedef __attribute__((ext_vector_type(8))) float v8f;

#define DIM 64
#define BN_EPS 1e-5f

// ---------------------------------------------------------------------------
// Degree / normalization
// ---------------------------------------------------------------------------
__global__ __launch_bounds__(256) void gcn_zero_i32(int* __restrict__ p, int n) {
  int i = blockIdx.x * 256 + threadIdx.x;
  if (i < n) p[i] = 0;
}

__global__ __launch_bounds__(256) void gcn_zero_f32(float* __restrict__ p, int n) {
  int i = blockIdx.x * blockDim.x + threadIdx.x;
  if (i < n) p[i] = 0.0f;
}

__global__ __launch_bounds__(256) void gcn_deg_count(const int* __restrict__ dst,
                                                     int* __restrict__ deg, int e) {
  int i = blockIdx.x * 256 + threadIdx.x;
  if (i < e) atomicAdd(&deg[dst[i]], 1);
}

__global__ __launch_bounds__(256) void gcn_make_dinv(const int* __restrict__ deg,
                                                     float* __restrict__ dinv, int n) {
  int i = blockIdx.x * 256 + threadIdx.x;
  if (i < n) dinv[i] = rsqrtf((float)(deg[i] + 1));  // +1 self-loop, always >= 1
}

// ---------------------------------------------------------------------------
// Dense GEMM  xw = z @ W   (N x 64) @ (64 x 64), fp32 via V_WMMA_F32_16X16X4_F32
// Block = 256 threads = 8 waves.  Each wave owns a full 16-row x 64-col stripe
// (4 accumulators), so every A fragment is loaded from global once and reused
// across 4 WMMAs.  Block covers 128 rows.  W staged in LDS with (k,k+1)
// row-pairing so each B fragment is one 8-byte LDS load.
// ---------------------------------------------------------------------------
__global__ __launch_bounds__(256)
void gcn_gemm64(const float* __restrict__ Z, const float* __restrict__ W,
                float* __restrict__ XW, int n) {
  __shared__ float Wsh[DIM * DIM];  // paired layout: [(k>>1)*128 + nn*2 + (k&1)]

  const int t = threadIdx.x;
#pragma unroll
  for (int i = 0; i < 16; ++i) {
    int idx = t + 256 * i;          // 0..4095
    int k = idx >> 6;
    int nn = idx & 63;
    Wsh[(k >> 1) * 128 + nn * 2 + (k & 1)] = W[idx];
  }
  __syncthreads();

  const int lane = t & 31;
  const int wv   = t >> 5;          // 0..7
  const int hh   = lane >> 4;       // lane half: 0 or 1
  const int mn   = lane & 15;       // m for A, n for B/C/D
  const int rowbase = blockIdx.x * 128 + wv * 16;

  int arow = rowbase + mn;
  if (arow > n - 1) arow = n - 1;   // clamp (no divergence; stores guarded below)
  const float* aptr = Z + (size_t)arow * DIM;

  v8f acc0 = {}, acc1 = {}, acc2 = {}, acc3 = {};
#pragma unroll
  for (int k0 = 0; k0 < DIM; k0 += 4) {
    const int ka = k0 + 2 * hh;     // even
    // A fragment: lanes 0-15 -> K = k0,k0+1 ; lanes 16-31 -> K = k0+2,k0+3
    const float2 av = *(const float2*)(aptr + ka);
    v2f a; a.x = av.x; a.y = av.y;
    const float* bbase = &Wsh[(ka >> 1) * 128 + mn * 2];
    // Four 16-col B fragments; each is one b64 LDS load (rows ka, ka+1)
    const float2 b0v = *(const float2*)(bbase + 0 * 32);
    const float2 b1v = *(const float2*)(bbase + 1 * 32);
    const float2 b2v = *(const float2*)(bbase + 2 * 32);
    const float2 b3v = *(const float2*)(bbase + 3 * 32);
    v2f b0; b0.x = b0v.x; b0.y = b0v.y;
    v2f b1; b1.x = b1v.x; b1.y = b1v.y;
    v2f b2; b2.x = b2v.x; b2.y = b2v.y;
    v2f b3; b3.x = b3v.x; b3.y = b3v.y;
    acc0 = __builtin_amdgcn_wmma_f32_16x16x4_f32(false, a, false, b0, (short)0, acc0, false, false);
    acc1 = __builtin_amdgcn_wmma_f32_16x16x4_f32(false, a, false, b1, (short)0, acc1, false, false);
    acc2 = __builtin_amdgcn_wmma_f32_16x16x4_f32(false, a, false, b2, (short)0, acc2, false, false);
    acc3 = __builtin_amdgcn_wmma_f32_16x16x4_f32(false, a, false, b3, (short)0, acc3, false, false);
  }

  // D layout: VGPR r -> row rowbase + r (+8 for lane half 1), col = 16*c + mn
#pragma unroll
  for (int r = 0; r < 8; ++r) {
    int row = rowbase + r + hh * 8;
    if (row < n) {
      float* orow = XW + (size_t)row * DIM + mn;
      orow[0]  = acc0[r];
      orow[16] = acc1[r];
      orow[32] = acc2[r];
      orow[48] = acc3[r];
    }
  }
}

// ---------------------------------------------------------------------------
// Aggregation:  agg[i] = dinv[i]^2 * xw[i] + bias   then edge scatter-adds
// ---------------------------------------------------------------------------
__global__ __launch_bounds__(256)
void gcn_agg_init(const float* __restrict__ xw, const float* __restrict__ dinv,
                  const float* __restrict__ bias, float* __restrict__ agg, int n) {
  int i = blockIdx.x * 256 + threadIdx.x;  // over n*16 float4 groups
  if (i >= n * 16) return;
  int node = i >> 4;
  int c4 = (i & 15) << 2;
  float w = dinv[node]; w = w * w;
  float4 v = *(const float4*)(xw + (size_t)node * DIM + c4);
  float4 o;
  o.x = v.x * w + bias[c4 + 0];
  o.y = v.y * w + bias[c4 + 1];
  o.z = v.z * w + bias[c4 + 2];
  o.w = v.w * w + bias[c4 + 3];
  *(float4*)(agg + (size_t)node * DIM + c4) = o;
}

__global__ __launch_bounds__(256)
void gcn_agg_edges(const float* __restrict__ xw, const int* __restrict__ src,
                   const int* __restrict__ dst, const float* __restrict__ dinv,
                   float* __restrict__ agg, int e) {
  int i = blockIdx.x * 256 + threadIdx.x;  // 16 lanes per edge
  int ed = i >> 4;
  if (ed >= e) return;
  int c4 = (i & 15) << 2;
  int s = src[ed];
  int d = dst[ed];
  float nm = dinv[s] * dinv[d];
  float4 v = *(const float4*)(xw + (size_t)s * DIM + c4);
  float* out = agg + (size_t)d * DIM + c4;
  atomicAdd(out + 0, v.x * nm);
  atomicAdd(out + 1, v.y * nm);
  atomicAdd(out + 2, v.z * nm);
  atomicAdd(out + 3, v.w * nm);
}

// ---------------------------------------------------------------------------
// BatchNorm (training-mode batch stats) + optional ReLU
// ---------------------------------------------------------------------------
__global__ __launch_bounds__(256)
void gcn_bn_stats(const float* __restrict__ agg, float* __restrict__ sums, int n) {
  __shared__ float ssum[256];
  __shared__ float ssq[256];
  const int col  = threadIdx.x & 63;
  const int rsub = threadIdx.x >> 6;  // 0..3
  const int row0 = blockIdx.x * 256;
  float s = 0.0f, q = 0.0f;
  for (int r = rsub; r < 256; r += 4) {
    int row = row0 + r;
    if (row < n) {
      float v = agg[(size_t)row * DIM + col];
      s += v;
      q += v * v;
    }
  }
  ssum[threadIdx.x] = s;
  ssq[threadIdx.x]  = q;
  __syncthreads();
  if (threadIdx.x < 64) {
    s = ssum[threadIdx.x] + ssum[threadIdx.x + 64] +
        ssum[threadIdx.x + 128] + ssum[threadIdx.x + 192];
    q = ssq[threadIdx.x] + ssq[threadIdx.x + 64] +
        ssq[threadIdx.x + 128] + ssq[threadIdx.x + 192];
    atomicAdd(&sums[col], s);
    atomicAdd(&sums[64 + col], q);
  }
}

__global__ void gcn_bn_finalize(float* __restrict__ sums, const float* __restrict__ g,
                                const float* __restrict__ be, float inv_n) {
  int c = threadIdx.x;  // 64 threads
  float mu  = sums[c] * inv_n;
  float var = sums[64 + c] * inv_n - mu * mu;
  float sc  = rsqrtf(var + BN_EPS) * g[c];
  sums[c]      = sc;               // scale
  sums[64 + c] = be[c] - mu * sc;  // shift
}

__global__ __launch_bounds__(256)
void gcn_bn_apply(const float* __restrict__ agg, const float* __restrict__ ss,
                  float* __restrict__ out, int total, int relu) {
  int i = blockIdx.x * 256 + threadIdx.x;
  if (i >= total) return;
  int c = i & 63;
  float v = agg[i] * ss[c] + ss[64 + c];
  if (relu) v = fmaxf(v, 0.0f);
  out[i] = v;
}

// ---------------------------------------------------------------------------
// Host launch
// ---------------------------------------------------------------------------
extern "C" void kernel_launch(void* const* d_in, const int* in_sizes, int n_in,
                              void* d_out, int out_size, void* d_ws, size_t ws_size,
                              hipStream_t stream) {
  (void)n_in; (void)out_size; (void)ws_size;

  const float* x  = (const float*)d_in[0];
  const int*   ei = (const int*)d_in[1];
  const int N = in_sizes[0] / DIM;
  const int E = in_sizes[1] / 2;
  const int* src = ei;
  const int* dst = ei + E;

  const float* Wl[3]  = {(const float*)d_in[2],  (const float*)d_in[6],  (const float*)d_in[10]};
  const float* bl[3]  = {(const float*)d_in[3],  (const float*)d_in[7],  (const float*)d_in[11]};
  const float* gl[3]  = {(const float*)d_in[4],  (const float*)d_in[8],  (const float*)d_in[12]};
  const float* bel[3] = {(const float*)d_in[5],  (const float*)d_in[9],  (const float*)d_in[13]};

  float* out = (float*)d_out;

  // Workspace layout (floats): dinv[N] | xw[N*64] | agg[N*64] | stats[128]
  float* ws    = (float*)d_ws;
  float* dinv  = ws;
  float* xw    = ws + N;
  float* agg   = xw + (size_t)N * DIM;
  float* stats = agg + (size_t)N * DIM;
  int*   deg   = (int*)agg;  // reuse agg region before its first real use

  // Degree + D^{-1/2}, computed once (identical for all 3 layers)
  gcn_zero_i32<<<(N + 255) / 256, 256, 0, stream>>>(deg, N);
  gcn_deg_count<<<(E + 255) / 256, 256, 0, stream>>>(dst, deg, E);
  gcn_make_dinv<<<(N + 255) / 256, 256, 0, stream>>>(deg, dinv, N);

  const float* zin = x;
  const float inv_n = 1.0f / (float)N;
  for (int l = 0; l < 3; ++l) {
    gcn_gemm64<<<(N + 127) / 128, 256, 0, stream>>>(zin, Wl[l], xw, N);
    gcn_agg_init<<<(N * 16 + 255) / 256, 256, 0, stream>>>(xw, dinv, bl[l], agg, N);
    gcn_agg_edges<<<(E * 16 + 255) / 256, 256, 0, stream>>>(xw, src, dst, dinv, agg, E);
    gcn_zero_f32<<<1, 128, 0, stream>>>(stats, 128);
    gcn_bn_stats<<<(N + 255) / 256, 256, 0, stream>>>(agg, stats, N);
    gcn_bn_finalize<<<1, 64, 0, stream>>>(stats, gl[l], bel[l], inv_n);
    gcn_bn_apply<<<(N * DIM + 255) / 256, 256, 0, stream>>>(
        agg, stats, out, N * DIM, (l < 2) ? 1 : 0);
    zin = out;  // next layer reads normalized activations from d_out
  }
}